// RendererTopK_32134945309178
// MI455X (gfx1250) — compile-verified
//
#include <hip/hip_runtime.h>

typedef __attribute__((ext_vector_type(2))) float v2f;
typedef __attribute__((ext_vector_type(8))) float v8f;

#define KTOP 10
#define MAXM 1024
#define IDXBITS 10u
#define IDXMASK ((1u << IDXBITS) - 1u)
// abs(q) bits with low IDXBITS cleared: sign bit dropped so tiny negative
// rounding noise around q~0 still ranks near the top (where it belongs).
#define QMASK 0x7FFFFC00u

__device__ __forceinline__ unsigned umin32(unsigned a, unsigned b) { return a < b ? a : b; }
__device__ __forceinline__ unsigned umax32(unsigned a, unsigned b) { return a > b ? a : b; }

// Branchless sorted-insert of key k into ascending a[0..KTOP-1] keeping the
// KTOP smallest. Uses the sorted invariant: new a[j] = clamp(k, a[j-1], a[j])
// (= median), new a[0] = min(a[0], k). Each slot is independent (reads only
// old values), and min(max(k,lo),hi) maps to fused v_maxmin_u32.
__device__ __forceinline__ void topk_insert(unsigned* a, unsigned k) {
  unsigned b0 = umin32(a[0], k);
#pragma unroll
  for (int j = KTOP - 1; j >= 1; --j) {
    a[j] = umin32(umax32(k, a[j - 1]), a[j]);
  }
  a[0] = b0;
}

// ---------------------------------------------------------------------------
// Kernel 1: per-gaussian prep. Invert 2x2 SPD covariance, fold mean into the
// rank-6 coefficient vector G[m][0..7] so that q(n,m) = P(n) . G(m) with
// P = (x^2, x*y, y^2, x, y, 1, 0, 0).
// ---------------------------------------------------------------------------
__global__ __launch_bounds__(256) void prep_gauss(const float* __restrict__ mus,
                                                  const float* __restrict__ covs,
                                                  float* __restrict__ G, int M) {
  int m = blockIdx.x * blockDim.x + threadIdx.x;
  if (m >= M) return;
  float mux = mus[2 * m + 0], muy = mus[2 * m + 1];
  float c00 = covs[4 * m + 0], c01 = covs[4 * m + 1];
  float c10 = covs[4 * m + 2], c11 = covs[4 * m + 3];
  float rdet = 1.0f / (c00 * c11 - c01 * c10);
  float ia = c11 * rdet, ib = -c01 * rdet, ic = c00 * rdet;
  float* g = G + 8 * m;
  g[0] = ia;
  g[1] = 2.0f * ib;
  g[2] = ic;
  g[3] = -2.0f * (ia * mux + ib * muy);
  g[4] = -2.0f * (ib * mux + ic * muy);
  g[5] = ia * mux * mux + 2.0f * ib * mux * muy + ic * muy * muy;
  g[6] = 0.0f;
  g[7] = 0.0f;
}

// ---------------------------------------------------------------------------
// Kernel 2: WMMA q-evaluation + branchless per-lane top-10 + blend.
// Block = 256 threads = 8 waves; each wave owns 16 points; block owns 128.
// D tile (16x16 f32): M-dim = gaussians, N-dim = points.
//   lane l: point = l&15, gaussian rows = 8*(l>>4) + r (r = VGPR 0..7).
// Unrolled by 2 tiles per iteration; high occupancy (~56 VGPRs, 32KB LDS)
// hides the per-iteration ds_load latency.
// ---------------------------------------------------------------------------
__global__ __launch_bounds__(256) void render_topk(const float* __restrict__ x,
                                                   const float* __restrict__ G,
                                                   const float* __restrict__ cols,
                                                   float* __restrict__ out,
                                                   int M) {
  __shared__ __align__(16) float sG[MAXM * 8];  // 32 KB

  {
    const float4* Gv = (const float4*)G;
    float4* sGv = (float4*)sG;
    for (int i = threadIdx.x; i < M * 2; i += 256) sGv[i] = Gv[i];
  }
  __syncthreads();

  const int lane  = threadIdx.x & 31;
  const int wave  = threadIdx.x >> 5;
  const int col   = lane & 15;   // point slot within wave / D-tile column
  const int khalf = lane >> 4;   // which K-pair this lane carries
  const int point = blockIdx.x * 128 + wave * 16 + col;

  // Loop-invariant B operand: this point's feature vector, per the 32-bit
  // 4x16 B layout (lanes 0-15: K=0,1 ; lanes 16-31: K=2,3).
  float px = x[2 * point + 0];
  float py = x[2 * point + 1];
  float f0 = px * px, f1 = px * py, f2 = py * py;
  v2f bLo, bHi;
  if (khalf == 0) { bLo.x = f0; bLo.y = f1; bHi.x = py;   bHi.y = 1.0f; }
  else            { bLo.x = f2; bLo.y = px; bHi.x = 0.0f; bHi.y = 0.0f; }

  // Per-lane top-10 as packed (|q|-bits | gaussian-index) keys, ascending.
  unsigned keys[KTOP];
#pragma unroll
  for (int t = 0; t < KTOP; ++t) keys[t] = 0xFFFFFFFFu;

  const int ntile = M >> 4;               // even (M multiple of 32)
  const int arow  = col * 8 + 2 * khalf;  // lane's A-slice offset in a row block

  // Two A-operand buffers, rotated across an unroll-by-2 loop.
  v2f aLo0 = *reinterpret_cast<const v2f*>(sG + arow);
  v2f aHi0 = *reinterpret_cast<const v2f*>(sG + arow + 4);
  v2f aLo1 = *reinterpret_cast<const v2f*>(sG + 128 + arow);
  v2f aHi1 = *reinterpret_cast<const v2f*>(sG + 128 + arow + 4);

  for (int t = 0; t < ntile; t += 2) {
    // ---- half-body A: process tile t, prefetch tile t+2 into buffer 0 ----
    {
      int tn = (t + 2 < ntile) ? (t + 2) : 0;  // wrap: harmless extra load
      const float* gn = sG + tn * 128 + arow;
      v2f nLo = *reinterpret_cast<const v2f*>(gn);
      v2f nHi = *reinterpret_cast<const v2f*>(gn + 4);

      v8f c = {};
      c = __builtin_amdgcn_wmma_f32_16x16x4_f32(false, aLo0, false, bLo,
                                                (short)0, c, false, false);
      c = __builtin_amdgcn_wmma_f32_16x16x4_f32(false, aHi0, false, bHi,
                                                (short)0, c, false, false);

      const unsigned gb = (unsigned)(t * 16 + khalf * 8);
#pragma unroll
      for (int r = 0; r < 8; ++r) {
        unsigned key = (__float_as_uint(c[r]) & QMASK) | gb | (unsigned)r;
        topk_insert(keys, key);
      }
      aLo0 = nLo;
      aHi0 = nHi;
    }
    // ---- half-body B: process tile t+1, prefetch tile t+3 into buffer 1 ----
    {
      int tn = (t + 3 < ntile) ? (t + 3) : 0;
      const float* gn = sG + tn * 128 + arow;
      v2f nLo = *reinterpret_cast<const v2f*>(gn);
      v2f nHi = *reinterpret_cast<const v2f*>(gn + 4);

      v8f c = {};
      c = __builtin_amdgcn_wmma_f32_16x16x4_f32(false, aLo1, false, bLo,
                                                (short)0, c, false, false);
      c = __builtin_amdgcn_wmma_f32_16x16x4_f32(false, aHi1, false, bHi,
                                                (short)0, c, false, false);

      const unsigned gb = (unsigned)((t + 1) * 16 + khalf * 8);
#pragma unroll
      for (int r = 0; r < 8; ++r) {
        unsigned key = (__float_as_uint(c[r]) & QMASK) | gb | (unsigned)r;
        topk_insert(keys, key);
      }
      aLo1 = nLo;
      aHi1 = nHi;
    }
  }

  // Merge with partner lane (same point, other gaussian half): snapshot the
  // partner's sorted list, then insert each of its keys into ours.
  unsigned ok[KTOP];
#pragma unroll
  for (int t = 0; t < KTOP; ++t)
    ok[t] = (unsigned)__shfl_xor((int)keys[t], 16, 32);
#pragma unroll
  for (int t = 0; t < KTOP; ++t) topk_insert(keys, ok[t]);

  // Lanes 0-15 finalize: recompute exact q per winner, exp, blend, normalize.
  if (khalf == 0) {
    float wsum = 0.0f, cr = 0.0f, cg = 0.0f, cb = 0.0f;
#pragma unroll
    for (int t = 0; t < KTOP; ++t) {
      int gi = (int)(keys[t] & IDXMASK);
      const float* g = sG + gi * 8;
      float q = fmaf(f0, g[0],
                fmaf(f1, g[1],
                fmaf(f2, g[2],
                fmaf(px, g[3],
                fmaf(py, g[4], g[5])))));
      float v = __expf(-0.5f * fabsf(q));  // |.| is a free src modifier
      const float* cc = cols + 3 * gi;
      wsum += v;
      cr = fmaf(v, cc[0], cr);
      cg = fmaf(v, cc[1], cg);
      cb = fmaf(v, cc[2], cb);
    }
    float s = 1.0f / (wsum + 1e-6f);
    out[3 * point + 0] = cr * s;
    out[3 * point + 1] = cg * s;
    out[3 * point + 2] = cb * s;
  }
}

// ---------------------------------------------------------------------------
extern "C" void kernel_launch(void* const* d_in, const int* in_sizes, int n_in,
                              void* d_out, int out_size, void* d_ws, size_t ws_size,
                              hipStream_t stream) {
  const float* x    = (const float*)d_in[0];  // [N, 2]
  const float* mus  = (const float*)d_in[1];  // [1, M, 2]
  const float* covs = (const float*)d_in[2];  // [1, M, 2, 2]
  const float* cols = (const float*)d_in[3];  // [1, M, 3]
  float* out = (float*)d_out;                 // [N, 3]

  int N = in_sizes[0] / 2;
  int M = in_sizes[1] / 2;

  float* G = (float*)d_ws;  // [M, 8] coefficient matrix, 32 KB

  prep_gauss<<<(M + 255) / 256, 256, 0, stream>>>(mus, covs, G, M);
  render_topk<<<N / 128, 256, 0, stream>>>(x, G, cols, out, M);
}